// SpatialTransformer_31344671326220
// MI455X (gfx1250) — compile-verified
//
#include <hip/hip_runtime.h>

// 3D spatial transformer (identity grid + displacement field, trilinear interp).
// vol: [D,H,W,4] f32, df: [D,H,W,3] f32, out: [D,H,W,4] f32.
//
// Roofline: ~216 MB unique traffic (df 59MB + vol 78.6MB + out 78.6MB) at
// 23.3 TB/s -> ~9.3us; ~0.6 GFLOP is negligible -> pure bandwidth/gather
// kernel. WMMA is inapplicable (per-voxel gathered corner operand, not a
// shared-operand matmul). Optimize the memory path instead:
//   - launch geometry == tensor geometry -> zero integer div/mod in the kernel
//   - coalesced global_load_b96 df reads, 8x global_load_b128 corner gathers
//     (C=4 floats == one 16B vector), global_store_b128 output
//   - gfx1250 global_prefetch_b8 to stream df and warm L2 for the
//     near-identity gathers (df ~ N(0,1) -> samples stay within a few voxels)

namespace {

constexpr int Dd = 160;
constexpr int Hh = 192;
constexpr int Ww = 160;
constexpr int YPB = 2;  // y-rows per block -> 320 threads = 10 wave32s

__global__ __launch_bounds__(Ww * YPB) void st3d_trilinear(
    const float* __restrict__ vol,   // [D,H,W,4]
    const float* __restrict__ df,    // [D,H,W,3]
    float* __restrict__ out)         // [D,H,W,4]
{
  // No division anywhere: geometry comes straight from the launch config.
  int x = threadIdx.x;                      // 0..W-1
  int y = blockIdx.x * YPB + threadIdx.y;   // 0..H-1
  int z = blockIdx.y;                       // 0..D-1
  int gid = (z * Hh + y) * Ww + x;          // voxel index, x fastest

  const float* dfp = df + (size_t)gid * 3;

  // CDNA5 prefetch path (global_prefetch_b8, speculative -> OOB dropped):
  // stream df ~24KB ahead; warm L2 along the volume's streaming direction.
  __builtin_prefetch(dfp + 3 * 2048, 0, 1);
  __builtin_prefetch(vol + (size_t)gid * 4 + (size_t)4 * Ww * 8, 0, 1);

  // Absolute sample location = identity grid + displacement.
  float lz = (float)z + dfp[0];
  float ly = (float)y + dfp[1];
  float lx = (float)x + dfp[2];

  // floor -> clamp corner0; corner1 = clamp(corner0 + 1)  (reference-exact).
  float fz = floorf(lz), fy = floorf(ly), fx = floorf(lx);
  float z0f = fminf(fmaxf(fz, 0.0f), (float)(Dd - 1));
  float y0f = fminf(fmaxf(fy, 0.0f), (float)(Hh - 1));
  float x0f = fminf(fmaxf(fx, 0.0f), (float)(Ww - 1));
  float z1f = fminf(z0f + 1.0f, (float)(Dd - 1));
  float y1f = fminf(y0f + 1.0f, (float)(Hh - 1));
  float x1f = fminf(x0f + 1.0f, (float)(Ww - 1));

  // corner0 weight = loc1_clamped - loc; corner1 weight = 1 - that.
  float wz0 = z1f - lz, wz1 = 1.0f - wz0;
  float wy0 = y1f - ly, wy1 = 1.0f - wy0;
  float wx0 = x1f - lx, wx1 = 1.0f - wx0;

  int iz0 = (int)z0f, iz1 = (int)z1f;
  int iy0 = (int)y0f, iy1 = (int)y1f;
  int ix0 = (int)x0f, ix1 = (int)x1f;

  // Each voxel's 4 channels are one 16-byte float4 -> global_load_b128 gather.
  const float4* __restrict__ v4 = reinterpret_cast<const float4*>(vol);
  int r00 = (iz0 * Hh + iy0) * Ww;
  int r01 = (iz0 * Hh + iy1) * Ww;
  int r10 = (iz1 * Hh + iy0) * Ww;
  int r11 = (iz1 * Hh + iy1) * Ww;

  float ax = 0.0f, ay = 0.0f, az = 0.0f, aw = 0.0f;

#define CORNER(ROW, IX, WZ, WY, WX)          \
  {                                          \
    float4 c = v4[(ROW) + (IX)];             \
    float w = (WZ) * (WY) * (WX);            \
    ax = fmaf(w, c.x, ax);                   \
    ay = fmaf(w, c.y, ay);                   \
    az = fmaf(w, c.z, az);                   \
    aw = fmaf(w, c.w, aw);                   \
  }

  // itertools.product corner order (matches reference accumulation order).
  CORNER(r00, ix0, wz0, wy0, wx0)
  CORNER(r00, ix1, wz0, wy0, wx1)
  CORNER(r01, ix0, wz0, wy1, wx0)
  CORNER(r01, ix1, wz0, wy1, wx1)
  CORNER(r10, ix0, wz1, wy0, wx0)
  CORNER(r10, ix1, wz1, wy0, wx1)
  CORNER(r11, ix0, wz1, wy1, wx0)
  CORNER(r11, ix1, wz1, wy1, wx1)
#undef CORNER

  float4 o;
  o.x = ax; o.y = ay; o.z = az; o.w = aw;
  reinterpret_cast<float4*>(out)[gid] = o;   // global_store_b128, coalesced
}

} // namespace

extern "C" void kernel_launch(void* const* d_in, const int* in_sizes, int n_in,
                              void* d_out, int out_size, void* d_ws, size_t ws_size,
                              hipStream_t stream) {
  const float* vol = (const float*)d_in[0];  // D*H*W*4 floats
  const float* df  = (const float*)d_in[1];  // D*H*W*3 floats
  float* out = (float*)d_out;                // D*H*W*4 floats

  dim3 block(Ww, YPB, 1);                    // 320 threads = 10 wave32s
  dim3 grid(Hh / YPB, Dd, 1);                // 96 x 160 blocks
  st3d_trilinear<<<grid, block, 0, stream>>>(vol, df, out);
}